// HierarchicalRegimeController_21526376087905
// MI455X (gfx1250) — compile-verified
//
#include <hip/hip_runtime.h>
#include <stdint.h>

// ---------------- problem sizes ----------------
#define B_ 64
#define T_ 2048
#define D_ 512
#define H_ 512
#define G_ 4
#define S_ 6
#define A_ 4
// TAU = 0.5  -> multiply logits+noise by 2.0f

// ---------------- workspace float offsets ----------------
enum : int {
  OFF_P    = 0,         // 64*16*512 T-partials
  OFF_MEAN = 524288,    // (B,D)
  OFF_DIFF = 557056,    // (B,D)
  OFF_HG   = 589824,    // pre-activations (B,H)
  OFF_HS   = 622592,
  OFF_HA   = 655360,
  OFF_AG   = 688128,    // activations (B,H)
  OFF_AS   = 720896,
  OFF_AA   = 753664,
  OFF_SLOG = 786432,    // (B, D*S) = 196608
  OFF_ALOG = 983040,    // (B, D*A) = 131072
  OFF_GLOG = 1114112,   // (B, G)   = 256
  OFF_PSEC = 1114368,   // 128 KL partials
  OFF_PAST = 1114496,   // 128 KL partials
};

// ---------------- output float offsets (tuple order, flattened) -------------
enum : int {
  O_GP  = 0,        // global_probs (64,4)
  O_SP  = 256,      // sector_probs (64,512,6)
  O_AP  = 196864,   // asset_probs  (64,512,4)
  O_FG  = 327936,   // feature_gate (64,512)
  O_KLG = 360704,
  O_KLS = 360705,
  O_KLA = 360706,
};

typedef float v2f __attribute__((ext_vector_type(2)));
typedef float v8f __attribute__((ext_vector_type(8)));

__device__ __forceinline__ float gelu_exact(float x) {
  return 0.5f * x * (1.0f + erff(x * 0.7071067811865476f));
}

// ---------------- JAX threefry2x32 (partitionable stream) -------------------
__device__ __forceinline__ uint32_t rotl32d(uint32_t x, int d) {
  return (x << d) | (x >> (32 - d));
}
__device__ __forceinline__ void tf2x32(uint32_t k0, uint32_t k1,
                                       uint32_t c0, uint32_t c1,
                                       uint32_t& o0, uint32_t& o1) {
  uint32_t ks2 = k0 ^ k1 ^ 0x1BD11BDAu;
  uint32_t x0 = c0 + k0, x1 = c1 + k1;
#define TFR(r) { x0 += x1; x1 = rotl32d(x1, (r)); x1 ^= x0; }
  TFR(13) TFR(15) TFR(26) TFR(6)   x0 += k1;  x1 += ks2 + 1u;
  TFR(17) TFR(29) TFR(16) TFR(24)  x0 += ks2; x1 += k0 + 2u;
  TFR(13) TFR(15) TFR(26) TFR(6)   x0 += k0;  x1 += k1 + 3u;
  TFR(17) TFR(29) TFR(16) TFR(24)  x0 += k1;  x1 += ks2 + 4u;
  TFR(13) TFR(15) TFR(26) TFR(6)   x0 += ks2; x1 += k0 + 5u;
#undef TFR
  o0 = x0; o1 = x1;
}
// element idx of a 32-bit random_bits draw, then JAX uniform(tiny,1) -> gumbel
__device__ __forceinline__ float gumbel_elem(uint32_t k0, uint32_t k1, uint32_t idx) {
  uint32_t o0, o1;
  tf2x32(k0, k1, 0u, idx, o0, o1);
  uint32_t bits = o0 ^ o1;
  float f = __uint_as_float((bits >> 9) | 0x3f800000u) - 1.0f;  // [0,1)
  float u = fmaxf(1.17549435e-38f, f);
  return -logf(-logf(u));
}

// ============================================================================
// 1) Stream x: partial sums over T (16 chunks of 128), float2 per thread.
// grid (16, 64), block 256
__global__ void __launch_bounds__(256) k_reduce_x(const float* __restrict__ x,
                                                  float* __restrict__ P) {
  const int tc = blockIdx.x, b = blockIdx.y, tid = threadIdx.x;
  const float* px = x + (size_t)b * (T_ * D_) + (size_t)tc * 128 * D_ + tid * 2;
  float s0 = 0.f, s1 = 0.f;
#pragma unroll 4
  for (int t = 0; t < 128; ++t) {
    v2f v = *(const v2f*)px;
    s0 += v.x; s1 += v.y;
    px += D_;
  }
  float* pp = P + (size_t)(b * 16 + tc) * D_ + tid * 2;
  pp[0] = s0; pp[1] = s1;
}

// 2) finalize seq_mean and telescoped diff_summary. grid 128, block 256
__global__ void __launch_bounds__(256) k_finalize(const float* __restrict__ x,
                                                  const float* __restrict__ P,
                                                  float* __restrict__ mean,
                                                  float* __restrict__ diff) {
  const int gid = blockIdx.x * 256 + threadIdx.x;  // < 32768
  const int b = gid >> 9, d = gid & 511;
  float s = 0.f;
#pragma unroll
  for (int tc = 0; tc < 16; ++tc) s += P[(b * 16 + tc) * D_ + d];
  mean[gid] = s * (1.0f / (float)T_);
  const size_t base = (size_t)b * (T_ * D_);
  diff[gid] = (x[base + (size_t)(T_ - 1) * D_ + d] - x[base + d]) * (1.0f / (float)(T_ - 1));
}

// ============================================================================
// 3) Generic fp32 WMMA GEMM:  C(64,N) = A(64,512) @ W(512,N) + bias
// 8 waves/block: wave = (mtile 0..3) x (2 local ntiles). grid.x = N/32.
__global__ void __launch_bounds__(256) k_gemm_wmma(const float* __restrict__ A,
                                                   const float* __restrict__ W,
                                                   const float* __restrict__ bias,
                                                   float* __restrict__ C, int N) {
  const int K = D_;
  const int tid = threadIdx.x;
  const int wave = tid >> 5, lane = tid & 31;
  const int lane16 = lane & 15, hi = lane >> 4;
  const int m0 = (wave & 3) << 4;
  const int n0 = ((blockIdx.x << 1) + (wave >> 2)) << 4;

  const float* Ar = A + (m0 + lane16) * K + (hi << 1);  // A[m][2*hi + v]
  const float* Wc = W + (n0 + lane16);                  // W[2*hi + v][n]
  v8f acc = {0.f, 0.f, 0.f, 0.f, 0.f, 0.f, 0.f, 0.f};

  for (int k = 0; k < K; k += 16) {
#pragma unroll
    for (int kk = 0; kk < 16; kk += 4) {
      const int kb = k + kk + (hi << 1);
      v2f a = *(const v2f*)(Ar + k + kk);
      v2f b;
      b.x = Wc[(size_t)kb * N];
      b.y = Wc[(size_t)(kb + 1) * N];
      acc = __builtin_amdgcn_wmma_f32_16x16x4_f32(false, a, false, b,
                                                  (short)0, acc, false, false);
    }
  }
  const float bv = bias[n0 + lane16];
  const int row0 = m0 + (hi << 3);
#pragma unroll
  for (int v = 0; v < 8; ++v)
    C[(size_t)(row0 + v) * N + n0 + lane16] = acc[v] + bv;
}

// ============================================================================
// 4) LayerNorm+GELU (global net) and GELU (sector/asset). grid 64, block 256
__global__ void __launch_bounds__(256) k_act(const float* __restrict__ hg,
                                             const float* __restrict__ hs,
                                             const float* __restrict__ ha,
                                             const float* __restrict__ lnw,
                                             const float* __restrict__ lnb,
                                             float* __restrict__ ag,
                                             float* __restrict__ as_,
                                             float* __restrict__ aa_) {
  __shared__ float red[256];
  const int b = blockIdx.x, tid = threadIdx.x;
  const int i0 = b * H_ + tid, i1 = i0 + 256;
  const float v0 = hg[i0], v1 = hg[i1];
  red[tid] = v0 + v1; __syncthreads();
  for (int s = 128; s > 0; s >>= 1) { if (tid < s) red[tid] += red[tid + s]; __syncthreads(); }
  const float mu = red[0] * (1.0f / (float)H_);
  __syncthreads();
  const float c0 = v0 - mu, c1 = v1 - mu;
  red[tid] = c0 * c0 + c1 * c1; __syncthreads();
  for (int s = 128; s > 0; s >>= 1) { if (tid < s) red[tid] += red[tid + s]; __syncthreads(); }
  const float rstd = rsqrtf(red[0] * (1.0f / (float)H_) + 1e-5f);
  ag[i0] = gelu_exact(c0 * rstd * lnw[tid] + lnb[tid]);
  ag[i1] = gelu_exact(c1 * rstd * lnw[tid + 256] + lnb[tid + 256]);
  as_[i0] = gelu_exact(hs[i0]); as_[i1] = gelu_exact(hs[i1]);
  aa_[i0] = gelu_exact(ha[i0]); aa_[i1] = gelu_exact(ha[i1]);
}

// ============================================================================
// 5) global head: (64,512)@(512,4)+b. grid 64, block 256
__global__ void __launch_bounds__(256) k_global_head(const float* __restrict__ ag,
                                                     const float* __restrict__ gW2,
                                                     const float* __restrict__ gb2,
                                                     float* __restrict__ glog) {
  __shared__ float4 red[256];
  const int b = blockIdx.x, tid = threadIdx.x;
  float4 acc = {0.f, 0.f, 0.f, 0.f};
  for (int d = tid; d < H_; d += 256) {
    const float a = ag[b * H_ + d];
    const float4 w = *(const float4*)(gW2 + d * G_);
    acc.x += a * w.x; acc.y += a * w.y; acc.z += a * w.z; acc.w += a * w.w;
  }
  red[tid] = acc; __syncthreads();
  for (int s = 128; s > 0; s >>= 1) {
    if (tid < s) {
      red[tid].x += red[tid + s].x; red[tid].y += red[tid + s].y;
      red[tid].z += red[tid + s].z; red[tid].w += red[tid + s].w;
    }
    __syncthreads();
  }
  if (tid < G_) {
    const float* r = (const float*)&red[0];
    glog[b * G_ + tid] = r[tid] + gb2[tid];
  }
}

// 6) global gumbel-softmax + KL (mean over B). grid 1, block 64
__global__ void __launch_bounds__(64) k_global_gumbel(const float* __restrict__ glog,
                                                      const float* __restrict__ gprior,
                                                      float* __restrict__ gp_out,
                                                      float* __restrict__ klg_out,
                                                      uint32_t k0, uint32_t k1) {
  __shared__ float red[64];
  const int b = threadIdx.x;
  float l[G_], q[G_];
#pragma unroll
  for (int g = 0; g < G_; ++g) { l[g] = glog[b * G_ + g]; q[g] = gprior[g]; }
  float mq = -3.4e38f, ml = -3.4e38f;
#pragma unroll
  for (int g = 0; g < G_; ++g) { mq = fmaxf(mq, q[g]); ml = fmaxf(ml, l[g]); }
  float sq = 0.f, sl = 0.f;
#pragma unroll
  for (int g = 0; g < G_; ++g) { sq += expf(q[g] - mq); sl += expf(l[g] - ml); }
  const float lseq = mq + logf(sq), lsel = ml + logf(sl);
  float kl = 0.f;
#pragma unroll
  for (int g = 0; g < G_; ++g) {
    const float lp = l[g] - lsel;
    kl += expf(lp) * (lp - (q[g] - lseq));
  }
  float y[G_]; float my = -3.4e38f;
#pragma unroll
  for (int g = 0; g < G_; ++g) {
    y[g] = (l[g] + gumbel_elem(k0, k1, (uint32_t)(b * G_ + g))) * 2.0f;  // /TAU
    my = fmaxf(my, y[g]);
  }
  float sy = 0.f;
#pragma unroll
  for (int g = 0; g < G_; ++g) { y[g] = expf(y[g] - my); sy += y[g]; }
  const float inv = 1.0f / sy;
#pragma unroll
  for (int g = 0; g < G_; ++g) gp_out[b * G_ + g] = y[g] * inv;
  red[b] = kl; __syncthreads();
  for (int s = 32; s > 0; s >>= 1) { if (b < s) red[b] += red[b + s]; __syncthreads(); }
  if (b == 0) *klg_out = red[0] * (1.0f / (float)B_);
}

// 7) sector/asset gumbel-softmax + per-block KL partial. grid 128, block 256
template <int SN>
__global__ void __launch_bounds__(256) k_gsm(const float* __restrict__ logits,
                                             const float* __restrict__ prior,
                                             float* __restrict__ probs,
                                             float* __restrict__ partial,
                                             uint32_t k0, uint32_t k1) {
  __shared__ float red[256];
  const int tid = threadIdx.x;
  const int gid = blockIdx.x * 256 + tid;  // (b,f) flat, < 32768
  float l[SN], q[SN];
#pragma unroll
  for (int s = 0; s < SN; ++s) { l[s] = logits[gid * SN + s]; q[s] = prior[s]; }
  float mq = -3.4e38f, ml = -3.4e38f;
#pragma unroll
  for (int s = 0; s < SN; ++s) { mq = fmaxf(mq, q[s]); ml = fmaxf(ml, l[s]); }
  float sq = 0.f, sl = 0.f;
#pragma unroll
  for (int s = 0; s < SN; ++s) { sq += expf(q[s] - mq); sl += expf(l[s] - ml); }
  const float lseq = mq + logf(sq), lsel = ml + logf(sl);
  float kl = 0.f;
#pragma unroll
  for (int s = 0; s < SN; ++s) {
    const float lp = l[s] - lsel;
    kl += expf(lp) * (lp - (q[s] - lseq));
  }
  const uint32_t base = (uint32_t)gid * (uint32_t)SN;
  float y[SN]; float my = -3.4e38f;
#pragma unroll
  for (int s = 0; s < SN; ++s) {
    y[s] = (l[s] + gumbel_elem(k0, k1, base + (uint32_t)s)) * 2.0f;  // /TAU
    my = fmaxf(my, y[s]);
  }
  float sy = 0.f;
#pragma unroll
  for (int s = 0; s < SN; ++s) { y[s] = expf(y[s] - my); sy += y[s]; }
  const float inv = 1.0f / sy;
#pragma unroll
  for (int s = 0; s < SN; ++s) probs[gid * SN + s] = y[s] * inv;
  red[tid] = kl; __syncthreads();
  for (int s = 128; s > 0; s >>= 1) { if (tid < s) red[tid] += red[tid + s]; __syncthreads(); }
  if (tid == 0) partial[blockIdx.x] = red[0];
}

// 8) feature gate. grid 128, block 256
__global__ void __launch_bounds__(256) k_gates(const float* __restrict__ gp,
                                               const float* __restrict__ sp,
                                               const float* __restrict__ ap,
                                               const float* __restrict__ g2f,
                                               const float* __restrict__ sec_emb,
                                               const float* __restrict__ ast_emb,
                                               float* __restrict__ fg) {
  const int gid = blockIdx.x * 256 + threadIdx.x;  // < 32768
  const int b = gid >> 9, f = gid & 511;
  float acc = 0.f;
#pragma unroll
  for (int g = 0; g < G_; ++g) acc += gp[b * G_ + g] * g2f[g * D_ + f];
#pragma unroll
  for (int s = 0; s < S_; ++s) acc += sp[gid * S_ + s] * sec_emb[s * D_ + f];
#pragma unroll
  for (int a = 0; a < A_; ++a) acc += ap[gid * A_ + a] * ast_emb[a * D_ + f];
  fg[gid] = 1.0f / (1.0f + expf(-acc));
}

// 9) deterministic final KL reduction. grid 1, block 128
__global__ void __launch_bounds__(128) k_kl_reduce(const float* __restrict__ psec,
                                                   const float* __restrict__ past,
                                                   float* __restrict__ kls,
                                                   float* __restrict__ kla) {
  __shared__ float red[128];
  const int tid = threadIdx.x;
  red[tid] = psec[tid]; __syncthreads();
  for (int s = 64; s > 0; s >>= 1) { if (tid < s) red[tid] += red[tid + s]; __syncthreads(); }
  if (tid == 0) *kls = red[0] * (1.0f / 32768.0f);
  __syncthreads();
  red[tid] = past[tid]; __syncthreads();
  for (int s = 64; s > 0; s >>= 1) { if (tid < s) red[tid] += red[tid + s]; __syncthreads(); }
  if (tid == 0) *kla = red[0] * (1.0f / 32768.0f);
}

// ============================================================================
// host-side threefry (for jax.random.split(key(42), 3), partitionable stream)
static void host_tf(uint32_t k0, uint32_t k1, uint32_t c0, uint32_t c1,
                    uint32_t& o0, uint32_t& o1) {
  uint32_t ks2 = k0 ^ k1 ^ 0x1BD11BDAu;
  uint32_t x0 = c0 + k0, x1 = c1 + k1;
#define HTFR(r) { x0 += x1; x1 = (uint32_t)((x1 << (r)) | (x1 >> (32 - (r)))); x1 ^= x0; }
  HTFR(13) HTFR(15) HTFR(26) HTFR(6)   x0 += k1;  x1 += ks2 + 1u;
  HTFR(17) HTFR(29) HTFR(16) HTFR(24)  x0 += ks2; x1 += k0 + 2u;
  HTFR(13) HTFR(15) HTFR(26) HTFR(6)   x0 += k0;  x1 += k1 + 3u;
  HTFR(17) HTFR(29) HTFR(16) HTFR(24)  x0 += k1;  x1 += ks2 + 4u;
  HTFR(13) HTFR(15) HTFR(26) HTFR(6)   x0 += ks2; x1 += k0 + 5u;
#undef HTFR
  o0 = x0; o1 = x1;
}

extern "C" void kernel_launch(void* const* d_in, const int* in_sizes, int n_in,
                              void* d_out, int out_size, void* d_ws, size_t ws_size,
                              hipStream_t stream) {
  (void)in_sizes; (void)n_in; (void)out_size; (void)ws_size;
  const float* x      = (const float*)d_in[0];
  const float* gW1    = (const float*)d_in[1];
  const float* gb1    = (const float*)d_in[2];
  const float* g_ln_w = (const float*)d_in[3];
  const float* g_ln_b = (const float*)d_in[4];
  const float* gW2    = (const float*)d_in[5];
  const float* gb2    = (const float*)d_in[6];
  const float* sW1    = (const float*)d_in[7];
  const float* sb1    = (const float*)d_in[8];
  const float* sW2    = (const float*)d_in[9];
  const float* sb2    = (const float*)d_in[10];
  const float* aW1    = (const float*)d_in[11];
  const float* ab1    = (const float*)d_in[12];
  const float* aW2    = (const float*)d_in[13];
  const float* ab2    = (const float*)d_in[14];
  const float* g_pri  = (const float*)d_in[15];
  const float* s_pri  = (const float*)d_in[16];
  const float* a_pri  = (const float*)d_in[17];
  const float* g2f_W  = (const float*)d_in[18];
  const float* secE   = (const float*)d_in[19];
  const float* astE   = (const float*)d_in[20];

  float* ws  = (float*)d_ws;
  float* out = (float*)d_out;

  // jax.random.split(jax.random.key(42), 3): key i = threefry((0,42),(0,i))
  uint32_t gk[3][2];
  for (uint32_t i = 0; i < 3; ++i) host_tf(0u, 42u, 0u, i, gk[i][0], gk[i][1]);

  // 1-2: seq_mean + diff_summary
  k_reduce_x<<<dim3(16, 64), 256, 0, stream>>>(x, ws + OFF_P);
  k_finalize<<<128, 256, 0, stream>>>(x, ws + OFF_P, ws + OFF_MEAN, ws + OFF_DIFF);

  // 3: first-layer GEMMs (fp32 WMMA)
  k_gemm_wmma<<<16, 256, 0, stream>>>(ws + OFF_MEAN, gW1, gb1, ws + OFF_HG, H_);
  k_gemm_wmma<<<16, 256, 0, stream>>>(ws + OFF_MEAN, sW1, sb1, ws + OFF_HS, H_);
  k_gemm_wmma<<<16, 256, 0, stream>>>(ws + OFF_DIFF, aW1, ab1, ws + OFF_HA, H_);

  // 4: LN+GELU / GELU
  k_act<<<64, 256, 0, stream>>>(ws + OFF_HG, ws + OFF_HS, ws + OFF_HA,
                                g_ln_w, g_ln_b, ws + OFF_AG, ws + OFF_AS, ws + OFF_AA);

  // 5: second-layer GEMMs
  k_gemm_wmma<<<96, 256, 0, stream>>>(ws + OFF_AS, sW2, sb2, ws + OFF_SLOG, D_ * S_);
  k_gemm_wmma<<<64, 256, 0, stream>>>(ws + OFF_AA, aW2, ab2, ws + OFF_ALOG, D_ * A_);
  k_global_head<<<64, 256, 0, stream>>>(ws + OFF_AG, gW2, gb2, ws + OFF_GLOG);

  // 6-7: gumbel-softmax + KL
  k_global_gumbel<<<1, 64, 0, stream>>>(ws + OFF_GLOG, g_pri, out + O_GP,
                                        out + O_KLG, gk[0][0], gk[0][1]);
  k_gsm<S_><<<128, 256, 0, stream>>>(ws + OFF_SLOG, s_pri, out + O_SP,
                                     ws + OFF_PSEC, gk[1][0], gk[1][1]);
  k_gsm<A_><<<128, 256, 0, stream>>>(ws + OFF_ALOG, a_pri, out + O_AP,
                                     ws + OFF_PAST, gk[2][0], gk[2][1]);

  // 8: feature gate
  k_gates<<<128, 256, 0, stream>>>(out + O_GP, out + O_SP, out + O_AP,
                                   g2f_W, secE, astE, out + O_FG);

  // 9: deterministic KL means
  k_kl_reduce<<<1, 128, 0, stream>>>(ws + OFF_PSEC, ws + OFF_PAST,
                                     out + O_KLS, out + O_KLA);
}